// GAT_dialoggcn_v1_20383914787442
// MI455X (gfx1250) — compile-verified
//
#include <hip/hip_runtime.h>
#include <hip/hip_bf16.h>

#define B_ 64
#define N_ 2048
#define H_ 256
#define NEGC 1e30f

typedef __attribute__((ext_vector_type(2))) float v2f;
typedef __attribute__((ext_vector_type(4))) float v4f;
typedef __attribute__((ext_vector_type(8))) float v8f;

__device__ __forceinline__ float wave_sum(float v) {
#pragma unroll
    for (int off = 16; off > 0; off >>= 1) v += __shfl_xor(v, off, 32);
    return v;
}
__device__ __forceinline__ float wave_max(float v) {
#pragma unroll
    for (int off = 16; off > 0; off >>= 1) v = fmaxf(v, __shfl_xor(v, off, 32));
    return v;
}

// Kernel 1: per-batch alpha = Q.wq + K.wk + b, mask, softmax; emit attn_weight,
// ws0 = w*sm, ws1 = w*(1-sm). Also zero U accumulator for kernel 2.
// One block per b, 256 threads (8 waves), wave-per-row dot over H=256.
__global__ __launch_bounds__(256) void k_alpha_softmax(
    const float* __restrict__ Q, const float* __restrict__ K,
    const int* __restrict__ adj, const int* __restrict__ s_mask,
    const float* __restrict__ w_lin, const float* __restrict__ b_lin,
    float* __restrict__ attn_w, float* __restrict__ ws0,
    float* __restrict__ ws1, float* __restrict__ U) {
    __shared__ float alpha[N_];
    __shared__ float red[8];
    const int b = blockIdx.x;
    const int t = threadIdx.x;
    const int wave = t >> 5, lane = t & 31;

    // zero U[b, 0:512] for kernel-2 atomics (deterministic per launch)
    U[b * 512 + t] = 0.0f;
    U[b * 512 + 256 + t] = 0.0f;

    // per-lane slices of w_k (h = lane*4 + {0..3} and +128)
    const v4f wk0 = *reinterpret_cast<const v4f*>(w_lin + H_ + lane * 4);
    const v4f wk1 = *reinterpret_cast<const v4f*>(w_lin + H_ + 128 + lane * 4);

    const float* Kb = K + (size_t)b * N_ * H_;
    for (int n = wave; n < N_; n += 8) {
        const float* row = Kb + (size_t)n * H_;
        v4f k0 = __builtin_nontemporal_load(reinterpret_cast<const v4f*>(row + lane * 4));
        v4f k1 = __builtin_nontemporal_load(reinterpret_cast<const v4f*>(row + 128 + lane * 4));
        float p = k0.x * wk0.x + k0.y * wk0.y + k0.z * wk0.z + k0.w * wk0.w +
                  k1.x * wk1.x + k1.y * wk1.y + k1.z * wk1.z + k1.w * wk1.w;
        p = wave_sum(p);
        if (lane == 0) alpha[n] = p;
    }

    // qbase = Q[b]·w_q + b_lin (block reduce over 256 threads)
    float qp = Q[b * H_ + t] * w_lin[t];
    qp = wave_sum(qp);
    if (lane == 0) red[wave] = qp;
    __syncthreads();   // alpha[] and red[] complete
    float qbase = b_lin[0];
#pragma unroll
    for (int i = 0; i < 8; ++i) qbase += red[i];
    __syncthreads();   // allow red[] reuse

    // masked logits, 8 rows per thread
    float a[8];
    float mx = -NEGC;
#pragma unroll
    for (int i = 0; i < 8; ++i) {
        const int n = t + 256 * i;
        const float m = (float)adj[b * N_ + n];
        a[i] = alpha[n] + qbase - (1.0f - m) * NEGC;
        mx = fmaxf(mx, a[i]);
    }
    mx = wave_max(mx);
    if (lane == 0) red[wave] = mx;
    __syncthreads();
#pragma unroll
    for (int i = 0; i < 8; ++i) mx = fmaxf(mx, red[i]);
    __syncthreads();

    float s = 0.0f;
#pragma unroll
    for (int i = 0; i < 8; ++i) {
        a[i] = __expf(a[i] - mx);
        s += a[i];
    }
    s = wave_sum(s);
    if (lane == 0) red[wave] = s;
    __syncthreads();
    s = 0.0f;
#pragma unroll
    for (int i = 0; i < 8; ++i) s += red[i];
    const float inv = 1.0f / s;

#pragma unroll
    for (int i = 0; i < 8; ++i) {
        const int n = t + 256 * i;
        const float w = a[i] * inv;
        const float sm = (float)s_mask[b * N_ + n];
        attn_w[b * N_ + n] = w;
        const float w0 = w * sm;
        ws0[b * N_ + n] = w0;
        ws1[b * N_ + n] = w - w0;
    }
}

// Kernel 2: U[b, h]     += sum_n ws0[b,n] * V[b,n,h]
//           U[b, 256+h] += sum_n ws1[b,n] * V[b,n,h]
// Grid (8 n-chunks, 64 b), 256 threads = one thread per h column. Coalesced NT
// streaming of V (read exactly once -> keep out of caches).
__global__ __launch_bounds__(256) void k_weighted_v(
    const float* __restrict__ V, const float* __restrict__ ws0,
    const float* __restrict__ ws1, float* __restrict__ U) {
    __shared__ float s0[256], s1[256];
    const int b = blockIdx.y;
    const int n0 = blockIdx.x * 256;
    const int t = threadIdx.x;
    s0[t] = ws0[b * N_ + n0 + t];
    s1[t] = ws1[b * N_ + n0 + t];
    __syncthreads();

    const float* Vb = V + ((size_t)b * N_ + n0) * H_ + t;
    float a0 = 0.0f, a1 = 0.0f;
#pragma unroll 8
    for (int j = 0; j < 256; ++j) {
        const float v = __builtin_nontemporal_load(Vb + (size_t)j * H_);
        a0 = fmaf(s0[j], v, a0);
        a1 = fmaf(s1[j], v, a1);
    }
    atomicAdd(&U[b * 512 + t], a0);
    atomicAdd(&U[b * 512 + 256 + t], a1);
}

// Kernel 3: attn_sum = [u0|u1] @ [Wr0|Wr1]^T  via V_WMMA_F32_16X16X4_F32.
// M=64 (b), N=256 (o), K=512. Wave-per-16x16 tile; 64 tiles over 8 blocks x 8
// waves; 128 chained K=4 WMMA steps per tile. EXEC all-1s, no divergence.
__global__ __launch_bounds__(256) void k_gemm_wmma(
    const float* __restrict__ U, const float* __restrict__ Wr0,
    const float* __restrict__ Wr1, float* __restrict__ out) {
    const int wave = threadIdx.x >> 5;
    const int lane = threadIdx.x & 31;
    const int tile = blockIdx.x * 8 + wave;  // 0..63
    const int mt = tile >> 4;                // b-tile 0..3
    const int nt = tile & 15;                // o-tile 0..15
    const int half = lane >> 4;              // K sub-select per ISA A/B layout
    const int l15 = lane & 15;

    const float* Urow = U + (mt * 16 + l15) * 512;   // A: M = lane&15
    const float* W0row = Wr0 + (nt * 16 + l15) * 256; // B: N = lane&15
    const float* W1row = Wr1 + (nt * 16 + l15) * 256;

    v8f c = {};
    // K = 0..255 uses u0 & Wr0
#pragma unroll 4
    for (int k0 = 0; k0 < 256; k0 += 4) {
        const int kk = k0 + 2 * half;  // VGPR0 holds K={0,2}, VGPR1 K={1,3}
        v2f av = *reinterpret_cast<const v2f*>(Urow + kk);
        v2f bv = *reinterpret_cast<const v2f*>(W0row + kk);
        c = __builtin_amdgcn_wmma_f32_16x16x4_f32(false, av, false, bv,
                                                  (short)0, c, false, false);
    }
    // K = 256..511 uses u1 & Wr1
#pragma unroll 4
    for (int k0 = 0; k0 < 256; k0 += 4) {
        const int kk = k0 + 2 * half;
        v2f av = *reinterpret_cast<const v2f*>(Urow + 256 + kk);
        v2f bv = *reinterpret_cast<const v2f*>(W1row + kk);
        c = __builtin_amdgcn_wmma_f32_16x16x4_f32(false, av, false, bv,
                                                  (short)0, c, false, false);
    }

    // C/D layout: VGPR r -> M = r + 8*half, N = lane&15
    float* ob = out + (mt * 16) * H_ + nt * 16;
#pragma unroll
    for (int r = 0; r < 8; ++r) {
        const int M = r + 8 * half;
        ob[M * H_ + l15] = c[r];
    }
}

extern "C" void kernel_launch(void* const* d_in, const int* in_sizes, int n_in,
                              void* d_out, int out_size, void* d_ws, size_t ws_size,
                              hipStream_t stream) {
    const float* Q = (const float*)d_in[0];
    const float* K = (const float*)d_in[1];
    const float* V = (const float*)d_in[2];
    const int* adj = (const int*)d_in[3];
    const int* s_mask = (const int*)d_in[4];
    const float* w_lin = (const float*)d_in[5];
    const float* b_lin = (const float*)d_in[6];
    const float* Wr0 = (const float*)d_in[7];
    const float* Wr1 = (const float*)d_in[8];

    float* out = (float*)d_out;
    float* attn_w = out;                  // (B,1,N) = 131072 floats
    float* attn_sum = out + B_ * N_;      // (B,H)  = 16384 floats

    float* ws0 = (float*)d_ws;            // B*N
    float* ws1 = ws0 + B_ * N_;           // B*N
    float* U = ws1 + B_ * N_;             // B*512

    k_alpha_softmax<<<B_, 256, 0, stream>>>(Q, K, adj, s_mask, w_lin, b_lin,
                                            attn_w, ws0, ws1, U);
    k_weighted_v<<<dim3(8, B_), 256, 0, stream>>>(V, ws0, ws1, U);
    k_gemm_wmma<<<8, 256, 0, stream>>>(U, Wr0, Wr1, attn_sum);
}